// CLDMoCo_46488726012439
// MI455X (gfx1250) — compile-verified
//
#include <hip/hip_runtime.h>
#include <math.h>
#include <stdint.h>

#define NPTS 262144
#define DIM  128
#define KCL  120
#define KP   128
#define PPB  2048

// WMMA kernels: 512 threads = 16 waves, 16 points/wave -> 256 points/block
#define BLK_THREADS 512
#define BLK_PTS     256

typedef __attribute__((ext_vector_type(2))) float v2f;
typedef __attribute__((ext_vector_type(8))) float v8f;

// ---------------------------------------------------------------------------
// CDNA5 async global->LDS copy (16B per lane), tracked by ASYNCcnt.
// GVS addressing: mem = SGPR base + 32-bit VGPR offset.  vdst holds LDS addr.
// ---------------------------------------------------------------------------
__device__ __forceinline__ void async_g2lds_b128(uint32_t lds_byte_addr,
                                                 uint32_t gbyte_off,
                                                 const void* gbase) {
  asm volatile("global_load_async_to_lds_b128 %0, %1, %2"
               :
               : "v"(lds_byte_addr), "v"(gbyte_off), "s"(gbase)
               : "memory");
}
__device__ __forceinline__ void wait_asynccnt0() {
  asm volatile("s_wait_asynccnt 0x0" ::: "memory");
}

// Stage KP*DIM floats (64 KB) of centroids into LDS with async copies.
__device__ __forceinline__ void stage_centroids(float* cs, const float* c, int tid) {
  uint32_t lbase = (uint32_t)(uintptr_t)cs + (uint32_t)tid * 16u;
  uint32_t goff  = (uint32_t)tid * 16u;
  const uint32_t step = BLK_THREADS * 16u;          // bytes per round
#pragma unroll
  for (uint32_t i = 0; i < (KP * DIM * 4u) / step; ++i)
    async_g2lds_b128(lbase + i * step, goff + i * step, (const void*)c);
  wait_asynccnt0();
  __syncthreads();
}

// ---------------------------------------------------------------------------
// init: c = x[:K] (rows K..KP-1 zeroed), zero loss accumulators
// ---------------------------------------------------------------------------
__global__ void cld_init_kernel(const float* __restrict__ f1, const float* __restrict__ f2,
                                float* __restrict__ c1, float* __restrict__ c2,
                                float* __restrict__ lacc) {
  int i = blockIdx.x * 256 + threadIdx.x;
  if (i < KP * DIM) {
    int k = i >> 7; // i / DIM
    c1[i] = (k < KCL) ? f1[i] : 0.0f;
    c2[i] = (k < KCL) ? f2[i] : 0.0f;
  }
  if (i < 8) lacc[i] = 0.0f;
}

// ---------------------------------------------------------------------------
// per-iteration prep: csq[k] = ||c_k||^2 (1e30 for pad rows), zero sums/counts
// grid = KP blocks of DIM threads
// ---------------------------------------------------------------------------
__global__ void cld_csq_clear_kernel(const float* __restrict__ c, float* __restrict__ csq,
                                     float* __restrict__ sums, float* __restrict__ counts) {
  __shared__ float red[DIM];
  int k = blockIdx.x, d = threadIdx.x;
  float v = c[k * DIM + d];
  red[d] = v * v;
  __syncthreads();
  for (int s = DIM / 2; s > 0; s >>= 1) {
    if (d < s) red[d] += red[d + s];
    __syncthreads();
  }
  if (d == 0) csq[k] = (k < KCL) ? red[0] : 1e30f;
  sums[k * DIM + d] = 0.0f;
  if (d == 0) counts[k] = 0.0f;
}

// ---------------------------------------------------------------------------
// assignment: per wave, 16 points x 128 clusters via V_WMMA_F32_16X16X4_F32,
// then argmin of (csq - 2*score) with lowest-index tie-break.
// ---------------------------------------------------------------------------
__global__ __launch_bounds__(BLK_THREADS) void cld_assign_kernel(const float* __restrict__ x,
                                                                 const float* __restrict__ c,
                                                                 const float* __restrict__ csq,
                                                                 int* __restrict__ cl) {
  __shared__ __align__(16) float cs[KP * DIM]; // 64 KB centroid stage
  int tid = threadIdx.x;
  stage_centroids(cs, c, tid);

  int lane  = tid & 31;
  int row16 = lane & 15;            // A row / B column within tile
  int kh    = (lane >> 4) << 1;     // k-subblock: 0 or 2
  int base_m = blockIdx.x * BLK_PTS + (tid >> 5) * 16;

  // A fragments: 16 points x 128 features, 32 steps of K=4
  const float* xrow = x + (base_m + row16) * DIM;
  v2f afrag[32];
#pragma unroll
  for (int s = 0; s < 32; ++s)
    afrag[s] = *(const v2f*)(xrow + 4 * s + kh);

  float best[8];
  int   bidx[8];
#pragma unroll
  for (int j = 0; j < 8; ++j) { best[j] = 3.0e38f; bidx[j] = 0x7FFFFFFF; }

  for (int t = 0; t < 8; ++t) {
    const float* crow = cs + (row16 + 16 * t) * DIM;
    v8f acc = {};
#pragma unroll
    for (int s = 0; s < 32; ++s) {
      v2f b = *(const v2f*)(crow + 4 * s + kh);
      acc = __builtin_amdgcn_wmma_f32_16x16x4_f32(false, afrag[s], false, b,
                                                  (short)0, acc, false, false);
    }
    int n = row16 + 16 * t;
    float cq = csq[n];
#pragma unroll
    for (int j = 0; j < 8; ++j) {
      float v = cq - 2.0f * acc[j];
      if (v < best[j] || (v == best[j] && n < bidx[j])) { best[j] = v; bidx[j] = n; }
    }
  }

  // reduce across the 16 lanes that share each point (xor stays within halves)
#pragma unroll
  for (int m = 1; m < 16; m <<= 1) {
#pragma unroll
    for (int j = 0; j < 8; ++j) {
      float ob = __shfl_xor(best[j], m, 32);
      int   oi = __shfl_xor(bidx[j], m, 32);
      if (ob < best[j] || (ob == best[j] && oi < bidx[j])) { best[j] = ob; bidx[j] = oi; }
    }
  }
  if (row16 == 0) {
    int pb = base_m + ((lane >> 4) << 3); // lane0 -> pts 0..7, lane16 -> pts 8..15
#pragma unroll
    for (int j = 0; j < 8; ++j) cl[pb + j] = bidx[j];
  }
}

// ---------------------------------------------------------------------------
// scatter-add: LDS-tiled partial sums (K x 32 slice), one atomic flush/block
// grid = (NPTS/PPB, 4), block = 256 (8 point-rows x 32 feature-cols)
// ---------------------------------------------------------------------------
__global__ void cld_accum_kernel(const float* __restrict__ x, const int* __restrict__ cl,
                                 float* __restrict__ sums, float* __restrict__ counts) {
  __shared__ float lsum[KCL * 32];
  __shared__ float lcnt[KCL];
  int tid = threadIdx.x;
  int col = tid & 31, row = tid >> 5;
  for (int i = tid; i < KCL * 32; i += 256) lsum[i] = 0.0f;
  if (tid < KCL) lcnt[tid] = 0.0f;
  __syncthreads();

  int dbase = blockIdx.y * 32;
  int pbase = blockIdx.x * PPB;
  bool do_cnt = (blockIdx.y == 0) && (col == 0);
  for (int p = row; p < PPB; p += 8) {
    int i = pbase + p;
    int k = cl[i];
    atomicAdd(&lsum[k * 32 + col], x[i * DIM + dbase + col]);
    if (do_cnt) atomicAdd(&lcnt[k], 1.0f);
  }
  __syncthreads();
  for (int i = tid; i < KCL * 32; i += 256)
    atomicAdd(&sums[(i >> 5) * DIM + dbase + (i & 31)], lsum[i]);
  if (blockIdx.y == 0 && tid < KCL) atomicAdd(&counts[tid], lcnt[tid]);
}

// ---------------------------------------------------------------------------
// centroid update: c = sums / max-with-1 count; pad rows -> 0
// ---------------------------------------------------------------------------
__global__ void cld_update_kernel(float* __restrict__ c, const float* __restrict__ sums,
                                  const float* __restrict__ counts) {
  int k = blockIdx.x, d = threadIdx.x;
  if (k < KCL) {
    float cnt = counts[k];
    float div = (cnt == 0.0f) ? 1.0f : cnt;
    c[k * DIM + d] = sums[k * DIM + d] / div;
  } else {
    c[k * DIM + d] = 0.0f;
  }
}

// ---------------------------------------------------------------------------
// cross-entropy: logits = (x . c)/T via WMMA, streaming logsumexp per point,
// label logit gathered in-pass; one atomicAdd per 8 points.
// ---------------------------------------------------------------------------
__global__ __launch_bounds__(BLK_THREADS) void cld_ce_kernel(const float* __restrict__ x,
                                                             const float* __restrict__ c,
                                                             const int* __restrict__ labels,
                                                             float* __restrict__ lacc) {
  __shared__ __align__(16) float cs[KP * DIM];
  int tid = threadIdx.x;
  stage_centroids(cs, c, tid);

  int lane  = tid & 31;
  int row16 = lane & 15;
  int kh    = (lane >> 4) << 1;
  int base_m = blockIdx.x * BLK_PTS + (tid >> 5) * 16;
  int mrow  = (lane >> 4) << 3;

  const float* xrow = x + (base_m + row16) * DIM;
  v2f afrag[32];
#pragma unroll
  for (int s = 0; s < 32; ++s)
    afrag[s] = *(const v2f*)(xrow + 4 * s + kh);

  int lab[8];
#pragma unroll
  for (int j = 0; j < 8; ++j) lab[j] = labels[base_m + mrow + j];

  float mx[8], ssum[8], ll[8];
#pragma unroll
  for (int j = 0; j < 8; ++j) { mx[j] = -3.0e38f; ssum[j] = 0.0f; ll[j] = 0.0f; }

  const float invT = 1.0f / 0.4f;
  for (int t = 0; t < 8; ++t) {
    const float* crow = cs + (row16 + 16 * t) * DIM;
    v8f acc = {};
#pragma unroll
    for (int s = 0; s < 32; ++s) {
      v2f b = *(const v2f*)(crow + 4 * s + kh);
      acc = __builtin_amdgcn_wmma_f32_16x16x4_f32(false, afrag[s], false, b,
                                                  (short)0, acc, false, false);
    }
    int n = row16 + 16 * t;
    bool valid = (n < KCL);
#pragma unroll
    for (int j = 0; j < 8; ++j) {
      float v = valid ? acc[j] * invT : -3.0e38f; // masked lanes contribute exp(-inf)=0
      if (n == lab[j]) ll[j] = v;
      float mn = fmaxf(mx[j], v);
      ssum[j] = ssum[j] * expf(mx[j] - mn) + expf(v - mn);
      mx[j] = mn;
    }
  }

  // merge (max,sum) pairs + label logit across the 16-lane halves
#pragma unroll
  for (int m = 1; m < 16; m <<= 1) {
#pragma unroll
    for (int j = 0; j < 8; ++j) {
      float om = __shfl_xor(mx[j], m, 32);
      float os = __shfl_xor(ssum[j], m, 32);
      float ol = __shfl_xor(ll[j], m, 32);
      float mn = fmaxf(mx[j], om);
      ssum[j] = ssum[j] * expf(mx[j] - mn) + os * expf(om - mn);
      mx[j] = mn;
      ll[j] += ol;
    }
  }
  if (row16 == 0) {
    float local = 0.0f;
#pragma unroll
    for (int j = 0; j < 8; ++j) local += mx[j] + logf(ssum[j]) - ll[j];
    atomicAdd(lacc, local);
  }
}

// ---------------------------------------------------------------------------
__global__ void cld_fin_kernel(const float* __restrict__ lacc, float* __restrict__ out) {
  out[0] = (lacc[0] + lacc[1]) / (2.0f * (float)NPTS);
}

// ---------------------------------------------------------------------------
extern "C" void kernel_launch(void* const* d_in, const int* in_sizes, int n_in,
                              void* d_out, int out_size, void* d_ws, size_t ws_size,
                              hipStream_t stream) {
  const float* f1 = (const float*)d_in[0];
  const float* f2 = (const float*)d_in[1];
  float* ws = (float*)d_ws;

  float* c1     = ws;                  // KP*DIM
  float* c2     = c1 + KP * DIM;       // KP*DIM
  float* csq    = c2 + KP * DIM;       // KP
  float* sums   = csq + KP;            // KP*DIM
  float* counts = sums + KP * DIM;     // KP
  float* lacc   = counts + KP;         // 8
  int*   cl1    = (int*)(lacc + 8);    // NPTS
  int*   cl2    = cl1 + NPTS;          // NPTS
  float* out    = (float*)d_out;

  cld_init_kernel<<<(KP * DIM + 255) / 256, 256, 0, stream>>>(f1, f2, c1, c2, lacc);

  for (int pass = 0; pass < 2; ++pass) {
    const float* x = pass ? f2 : f1;
    float* c = pass ? c2 : c1;
    int* cl  = pass ? cl2 : cl1;
    for (int it = 0; it < 5; ++it) {
      cld_csq_clear_kernel<<<KP, DIM, 0, stream>>>(c, csq, sums, counts);
      cld_assign_kernel<<<NPTS / BLK_PTS, BLK_THREADS, 0, stream>>>(x, c, csq, cl);
      cld_accum_kernel<<<dim3(NPTS / PPB, 4), 256, 0, stream>>>(x, cl, sums, counts);
      cld_update_kernel<<<KP, DIM, 0, stream>>>(c, sums, counts);
    }
  }

  cld_ce_kernel<<<NPTS / BLK_PTS, BLK_THREADS, 0, stream>>>(f1, c2, cl2, lacc + 0);
  cld_ce_kernel<<<NPTS / BLK_PTS, BLK_THREADS, 0, stream>>>(f2, c1, cl1, lacc + 1);
  cld_fin_kernel<<<1, 1, 0, stream>>>(lacc, out);
}